// loss_DBL_SR_SDMA_21827023798277
// MI455X (gfx1250) — compile-verified
//
#include <hip/hip_runtime.h>

// Problem constants (match reference)
constexpr int BATCH = 8192;
constexpr int NANT  = 64;   // antennas (contraction dim)
constexpr int KUSR  = 16;   // users (M and N of the 16x16 output)
constexpr float NOISE = 1.0f;

typedef __attribute__((ext_vector_type(2))) float v2f;  // A/B fragment: 16x4 / 4x16 f32
typedef __attribute__((ext_vector_type(8))) float v8f;  // C/D fragment: 16x16 f32

__global__ void zero_out_kernel(float* out) {
    if (threadIdx.x == 0) out[0] = 0.0f;
}

__global__ __launch_bounds__(256) void
sumrate_wmma_kernel(const float* __restrict__ H,   // [B, N, K, 2]
                    const float* __restrict__ P,   // [B, 2*N*K] (real block, imag block)
                    float* __restrict__ out)       // scalar
{
    const int wave = threadIdx.x >> 5;
    const int lane = threadIdx.x & 31;
    const int b    = blockIdx.x * 8 + wave;       // one sample per wave32

    const int row  = lane & 15;                   // M (for A) / N column (for B)
    const int half = lane >> 4;                   // 0: K=0,1  1: K=2,3 within a 4-chunk

    const float* __restrict__ Hb  = H + (size_t)b * (NANT * KUSR * 2);
    const float* __restrict__ Prb = P + (size_t)b * (2 * NANT * KUSR);
    const float* __restrict__ Pib = Prb + NANT * KUSR;

    v8f Dr = {};   // real part of D = H^T P
    v8f Di = {};   // imag part

    // 16 chunks of 4 along the antenna (contraction) dimension.
    // A fragment (16x4 f32): lane<16 -> {K=nb, K=nb+1}; lane>=16 -> {K=nb+2, nb+3}
    // A[M=row, K=n] = H[n, row];  B[K=n, N=row] = P[n, row]
#pragma unroll
    for (int c = 0; c < NANT / 4; ++c) {
        const int nb = 4 * c + 2 * half;

        // H[b, nb..nb+1, row, {re,im}] : float2 per n, stride K*2 floats between n
        float2 h0 = *(const float2*)(Hb + (size_t)nb       * (KUSR * 2) + row * 2);
        float2 h1 = *(const float2*)(Hb + (size_t)(nb + 1) * (KUSR * 2) + row * 2);

        v2f ar  = { h0.x,  h1.x};
        v2f ai  = { h0.y,  h1.y};
        v2f ain = {-h0.y, -h1.y};   // f32 WMMA NEG only allows C-negate -> negate in VALU

        v2f br = { Prb[nb * KUSR + row], Prb[(nb + 1) * KUSR + row] };
        v2f bi = { Pib[nb * KUSR + row], Pib[(nb + 1) * KUSR + row] };

        // Dr += Hr^T Pr - Hi^T Pi ; Di += Hr^T Pi + Hi^T Pr
        Dr = __builtin_amdgcn_wmma_f32_16x16x4_f32(false, ar,  false, br, (short)0, Dr, false, false);
        Dr = __builtin_amdgcn_wmma_f32_16x16x4_f32(false, ain, false, bi, (short)0, Dr, false, false);
        Di = __builtin_amdgcn_wmma_f32_16x16x4_f32(false, ar,  false, bi, (short)0, Di, false, false);
        Di = __builtin_amdgcn_wmma_f32_16x16x4_f32(false, ai,  false, br, (short)0, Di, false, false);
    }

    // Epilogue on the 16x16 C/D layout:
    //   VGPR r, lanes 0-15  -> row M=r,   col N=lane
    //   VGPR r, lanes 16-31 -> row M=r+8, col N=lane-16
    float rate = 0.0f;
#pragma unroll
    for (int r = 0; r < 8; ++r) {
        float m2 = Dr[r] * Dr[r] + Di[r] * Di[r];

        // diagonal (N == M): lane r for the low half, lane 24+r for the high half
        int srcLane = (lane < 16) ? r : (24 + r);
        float sig = __shfl(m2, srcLane, 32);

        // row total: butterfly within each 16-lane half (masks < 16 stay in-half)
        float tot = m2;
        tot += __shfl_xor(tot, 1, 32);
        tot += __shfl_xor(tot, 2, 32);
        tot += __shfl_xor(tot, 4, 32);
        tot += __shfl_xor(tot, 8, 32);

        float interf = tot - sig + NOISE;
        rate += __log2f(1.0f + sig / interf);
    }

    // lane 0 holds rows 0..7 summed, lane 16 holds rows 8..15 summed
    float rHi = __shfl(rate, 16, 32);
    if (lane == 0) {
        atomicAdd(out, -(rate + rHi) * (1.0f / (float)BATCH));
    }
}

extern "C" void kernel_launch(void* const* d_in, const int* in_sizes, int n_in,
                              void* d_out, int out_size, void* d_ws, size_t ws_size,
                              hipStream_t stream) {
    const float* H = (const float*)d_in[0];   // [B, N, K, 2] f32
    const float* P = (const float*)d_in[1];   // [B, 2*N*K]  f32
    float* out = (float*)d_out;               // scalar f32

    zero_out_kernel<<<1, 64, 0, stream>>>(out);
    sumrate_wmma_kernel<<<dim3(BATCH / 8), 256, 0, stream>>>(H, P, out);
}